// ChunkedTGnnModel_18124761989845
// MI455X (gfx1250) — compile-verified
//
#include <hip/hip_runtime.h>

// ChunkedTGnn on MI455X (gfx1250): the per-chunk tiled graph is the base graph
// replicated per timestep, so the model is 512 independent GCN-2layer passes:
//   Z = H@W ; out = relu(A_hat Z + b), twice.
// Dense H@W uses V_WMMA_F32_16X16X4_F32 (fp32 WMMA, wave32); each wave computes
// a 16x64 output strip (4 accumulators, 64 chained WMMAs). W is staged in LDS
// pre-swizzled in K-pairs so each B fragment is a single aligned ds_load_b64
// (no repacking movs). Epilogue fuses the GCN self-loop + bias init
// (out = b + dinv[n]^2 * Z). Sparse aggregation = global f32 atomics.

#define T_STEPS 512
#define D 64

typedef float v2f __attribute__((ext_vector_type(2)));
typedef float v8f __attribute__((ext_vector_type(8)));

__device__ __forceinline__ void atomAddF(float* p, float v) {
    __hip_atomic_fetch_add(p, v, __ATOMIC_RELAXED, __HIP_MEMORY_SCOPE_AGENT);
}

// ---------------- normalization precompute ----------------

__global__ void deg_init_k(float* __restrict__ deg, int N) {
    int i = blockIdx.x * blockDim.x + threadIdx.x;
    if (i < N) deg[i] = 1.0f;  // self loop
}

__global__ void deg_count_k(const int* __restrict__ col, float* __restrict__ deg, int E) {
    int e = blockIdx.x * blockDim.x + threadIdx.x;
    if (e < E) atomAddF(&deg[col[e]], 1.0f);  // exact for counts < 2^24
}

__global__ void dinv_k(float* __restrict__ deg, int N) {
    int i = blockIdx.x * blockDim.x + threadIdx.x;
    if (i < N) deg[i] = rsqrtf(deg[i]);  // deg >= 1 always
}

__global__ void enorm_k(const int* __restrict__ ei, const float* __restrict__ dinv,
                        float* __restrict__ enorm, int E) {
    int e = blockIdx.x * blockDim.x + threadIdx.x;
    if (e < E) enorm[e] = dinv[ei[e]] * dinv[ei[E + e]];
}

// ---------------- fused GEMM + self-loop init ----------------
// Z[M,64] = A[M,64] @ W[64,64]; out[m,:] = bias + dinv[m/T]^2 * Z[m,:]
// Block = 128 threads = 4 waves (wave32), 64 rows per block; wave w owns rows
// [m0, m0+16) and all 64 output columns (4 x v8f accumulators).
// A-frag (32-bit A 16x4): lanes 0-15 -> K{0,1}, lanes 16-31 -> K{2,3}, M=lane%16.
// B-frag mirrored over N. C/D: vgpr r = row (lane/16)*8+r, col lane%16.
// LDS W layout: sW[(k/2)*128 + n*2 + (k&1)] -> B pair (W[k][n],W[k+1][n]) is one
// aligned 8B word per lane -> single ds_load_b64 into an even VGPR pair.
// NOTE: safe when A == out (conv2): row m is read only by the wave that writes it.

__global__ void gemm64_wmma_fused_k(const float* __restrict__ A,
                                    const float* __restrict__ W,
                                    float* __restrict__ Z,
                                    const float* __restrict__ dinv,
                                    const float* __restrict__ bias,
                                    float* __restrict__ out) {
    __shared__ float sW[D * D];  // 16KB, K-pair swizzled
    const int tid  = threadIdx.x;
    const int wave = tid >> 5;
    const int lane = tid & 31;
    const int hi   = lane >> 4;   // 0: K pair {0,1}; 1: K pair {2,3}
    const int l16  = lane & 15;

    for (int i = tid; i < D * D; i += 128) {
        const int k = i >> 6, n = i & (D - 1);
        sW[(k >> 1) * 128 + n * 2 + (k & 1)] = W[i];
    }
    __syncthreads();

    const size_t m0 = ((size_t)blockIdx.x * 4 + wave) * 16;
    const float* __restrict__ Arow = A + m0 * D;

    // per-lane bias values for the 4 column tiles
    const float bb0 = bias[l16 +  0];
    const float bb1 = bias[l16 + 16];
    const float bb2 = bias[l16 + 32];
    const float bb3 = bias[l16 + 48];

    v8f c0 = {}, c1 = {}, c2 = {}, c3 = {};
    #pragma unroll
    for (int kk = 0; kk < 16; ++kk) {
        const int kbase = kk * 4 + hi * 2;        // even -> 8B aligned
        const int krow  = (kbase >> 1) * 128;     // swizzled LDS row of this K pair
        v2f a  = *(const v2f*)(Arow + l16 * D + kbase);
        v2f b0 = *(const v2f*)(sW + krow + (l16 +  0) * 2);
        v2f b1 = *(const v2f*)(sW + krow + (l16 + 16) * 2);
        v2f b2 = *(const v2f*)(sW + krow + (l16 + 32) * 2);
        v2f b3 = *(const v2f*)(sW + krow + (l16 + 48) * 2);
        c0 = __builtin_amdgcn_wmma_f32_16x16x4_f32(false, a, false, b0, (short)0, c0, false, false);
        c1 = __builtin_amdgcn_wmma_f32_16x16x4_f32(false, a, false, b1, (short)0, c1, false, false);
        c2 = __builtin_amdgcn_wmma_f32_16x16x4_f32(false, a, false, b2, (short)0, c2, false, false);
        c3 = __builtin_amdgcn_wmma_f32_16x16x4_f32(false, a, false, b3, (short)0, c3, false, false);
    }

    #pragma unroll
    for (int r = 0; r < 8; ++r) {
        const size_t m = m0 + (size_t)(hi * 8 + r);
        const float dn = dinv[m / T_STEPS];
        const float dn2 = dn * dn;
        const size_t base = m * D + l16;
        Z[base +  0] = c0[r];  out[base +  0] = bb0 + dn2 * c0[r];
        Z[base + 16] = c1[r];  out[base + 16] = bb1 + dn2 * c1[r];
        Z[base + 32] = c2[r];  out[base + 32] = bb2 + dn2 * c2[r];
        Z[base + 48] = c3[r];  out[base + 48] = bb3 + dn2 * c3[r];
    }
}

// ---------------- sparse aggregation ----------------
// out[col,t,f] += norm[e] * Z[row,t,f]  for every edge e, timestep t, feature f
__global__ void agg_scatter_k(const float* __restrict__ Z, const int* __restrict__ ei,
                              const float* __restrict__ enorm, float* __restrict__ out,
                              int E) {
    size_t i = (size_t)blockIdx.x * blockDim.x + threadIdx.x;
    size_t total = (size_t)E * T_STEPS * D;
    if (i >= total) return;
    const int f  = (int)(i & (D - 1));
    size_t et    = i >> 6;
    const int t  = (int)(et % T_STEPS);
    const int e  = (int)(et / T_STEPS);
    const int r  = ei[e];
    const int cc = ei[E + e];
    const float v = enorm[e] * Z[((size_t)r * T_STEPS + t) * D + f];
    atomAddF(out + ((size_t)cc * T_STEPS + t) * D + f, v);
}

__global__ void relu_k(float* __restrict__ out, size_t total) {
    size_t i = (size_t)blockIdx.x * blockDim.x + threadIdx.x;
    if (i < total) out[i] = fmaxf(out[i], 0.0f);
}

// ---------------- launch ----------------

extern "C" void kernel_launch(void* const* d_in, const int* in_sizes, int n_in,
                              void* d_out, int out_size, void* d_ws, size_t ws_size,
                              hipStream_t stream) {
    const float* x  = (const float*)d_in[0];  // [N, T, 64]
    const int*   ei = (const int*)d_in[1];    // [2, E]
    const float* W1 = (const float*)d_in[2];  // [64, 64]
    const float* b1 = (const float*)d_in[3];  // [64]
    const float* W2 = (const float*)d_in[4];  // [64, 64]
    const float* b2 = (const float*)d_in[5];  // [64]
    float* out = (float*)d_out;               // [N, T, 64]

    const int M = in_sizes[0] / D;     // N*T flattened rows
    const int N = M / T_STEPS;
    const int E = in_sizes[1] / 2;

    // workspace layout: dinv[N] | enorm[E] | Z[M*64]
    float* ws = (float*)d_ws;
    size_t off = 0;
    float* dinv = ws;                 off += ((size_t)N + 63) & ~(size_t)63;
    float* enrm = ws + off;           off += ((size_t)E + 63) & ~(size_t)63;
    float* Zbuf = ws + off;
    (void)ws_size; (void)n_in; (void)out_size;

    const int TB = 256;
    const size_t totNT = (size_t)M * D;
    const size_t totE  = (size_t)E * T_STEPS * D;
    const int gNT = (int)((totNT + TB - 1) / TB);
    const int gE  = (int)((totE + TB - 1) / TB);
    const int gM  = M / 64;  // M = 512000 = 64*8000 -> EXEC all-1s for WMMA

    // gcn_norm
    deg_init_k <<<(N + TB - 1) / TB, TB, 0, stream>>>(dinv, N);
    deg_count_k<<<(E + TB - 1) / TB, TB, 0, stream>>>(ei + E, dinv, E);
    dinv_k     <<<(N + TB - 1) / TB, TB, 0, stream>>>(dinv, N);
    enorm_k    <<<(E + TB - 1) / TB, TB, 0, stream>>>(ei, dinv, enrm, E);

    // conv1: Z = X@W1 ; out = b1 + selfloop*Z ; out += edge msgs ; relu
    gemm64_wmma_fused_k<<<gM, 128, 0, stream>>>(x, W1, Zbuf, dinv, b1, out);
    agg_scatter_k      <<<gE, TB, 0, stream>>>(Zbuf, ei, enrm, out, E);
    relu_k             <<<gNT, TB, 0, stream>>>(out, totNT);

    // conv2: Z = H1@W2 ; out = b2 + selfloop*Z ; out += edge msgs ; relu
    gemm64_wmma_fused_k<<<gM, 128, 0, stream>>>(out, W2, Zbuf, dinv, b2, out);
    agg_scatter_k      <<<gE, TB, 0, stream>>>(Zbuf, ei, enrm, out, E);
    relu_k             <<<gNT, TB, 0, stream>>>(out, totNT);
}